// ModelClass_39273180954932
// MI455X (gfx1250) — compile-verified
//
#include <hip/hip_runtime.h>
#include <math.h>

#define B_GRAPHS 64
#define NPG      2048
#define KNN      10
#define TILES    (NPG / 16)   // 128
#define THREADS  256
#define WGS_PER_GRAPH 16      // 16 WGs x 8 waves x 16 queries = 2048

#define IDX_BITS  11u
#define IDX_MASK  0x7FFu
#define KEY_MASK  0xFFFFF800u

typedef float v2f __attribute__((ext_vector_type(2)));
typedef float v8f __attribute__((ext_vector_type(8)));

// LDS layout:
//   [0      , 32768)  : quads  float4[2048]  (sx, sy, |s|^2, 1)  -- phase 0/1
//                       (lo 8B = A-operand for lanes 0-15, hi 8B = lanes 16-31)
//   [0      , 10240)  : topK   u32[8][2][16][10]                 -- phase 2 (aliased)
//   [32768  , 57344)  : hL     float[2048*3]                     -- all phases
#define SMEM_BYTES (32768 + 24576)

__global__ __launch_bounds__(THREADS)
void gravnet_knn_kernel(const float* __restrict__ x,
                        const float* __restrict__ Ws, const float* __restrict__ bs,
                        const float* __restrict__ Wh, const float* __restrict__ bh,
                        const float* __restrict__ Wo, const float* __restrict__ bo,
                        float* __restrict__ out)
{
    __shared__ __align__(16) unsigned char smemRaw[SMEM_BYTES];
    float4* quads = (float4*)smemRaw;                  // [NPG]
    float*  hL    = (float*)(smemRaw + 32768);         // [NPG*3]

    const int graph = blockIdx.x >> 4;
    const int wgIn  = blockIdx.x & (WGS_PER_GRAPH - 1);
    const int t     = threadIdx.x;
    const int gbase = graph * NPG;

    // ---------------- Phase 0: s-quads + h into LDS ----------------
    {
        const float ws00 = Ws[0], ws01 = Ws[1], ws02 = Ws[2];
        const float ws10 = Ws[3], ws11 = Ws[4], ws12 = Ws[5];
        const float b0 = bs[0], b1 = bs[1];
        const float wh00 = Wh[0], wh01 = Wh[1], wh02 = Wh[2];
        const float wh10 = Wh[3], wh11 = Wh[4], wh12 = Wh[5];
        const float wh20 = Wh[6], wh21 = Wh[7], wh22 = Wh[8];
        const float c0 = bh[0], c1 = bh[1], c2 = bh[2];
        for (int p = t; p < NPG; p += THREADS) {
            const float* xp = x + (size_t)(gbase + p) * 3;
            float x0 = xp[0], x1 = xp[1], x2 = xp[2];
            float s0 = fmaf(ws02, x2, fmaf(ws01, x1, fmaf(ws00, x0, b0)));
            float s1 = fmaf(ws12, x2, fmaf(ws11, x1, fmaf(ws10, x0, b1)));
            quads[p] = make_float4(s0, s1, fmaf(s0, s0, s1 * s1), 1.0f);
            hL[p * 3 + 0] = fmaf(wh02, x2, fmaf(wh01, x1, fmaf(wh00, x0, c0)));
            hL[p * 3 + 1] = fmaf(wh12, x2, fmaf(wh11, x1, fmaf(wh10, x0, c1)));
            hL[p * 3 + 2] = fmaf(wh22, x2, fmaf(wh21, x1, fmaf(wh20, x0, c2)));
        }
    }
    __syncthreads();

    // ---------------- Phase 1: WMMA distance tiles + packed-key top-K ----------------
    const int lane = t & 31;       // wave32
    const int wave = t >> 5;       // 8 waves
    const int col  = lane & 15;    // query column inside wave tile
    const int hi   = lane >> 4;    // lane half (K-split / row-split)

    // B operand (queries), invariant over the whole scan.
    // B 4x16 layout: lanes 0-15 hold K=0/K=1; lanes 16-31 hold K=2/K=3.
    const int qLocal = wgIn * 128 + wave * 16 + col;
    float4 q = quads[qLocal];
    v2f Bop;
    Bop.x = hi ? 1.0f : (-2.0f * q.x);   // K2 coeff vs -2*sx
    Bop.y = hi ? q.z  : (-2.0f * q.y);   // |s_i|^2 vs -2*sy

    // A operand comes straight from LDS: lanes 0-15 read (sx,sy), lanes 16-31 read (|s|^2,1).
    const v2f* apairs = (const v2f*)smemRaw;   // quads viewed as [NPG][2] float2

    unsigned topK[KNN];
#pragma unroll
    for (int k = 0; k < KNN; ++k) topK[k] = 0xFFFFFFFFu;

    const v8f Zc = {};   // zero accumulator

#pragma unroll 2
    for (int tile = 0; tile < TILES; ++tile) {
        const int j0 = tile * 16;
        v2f Aop = apairs[((j0 + col) << 1) | hi];   // one ds_load_b64

        v8f D = __builtin_amdgcn_wmma_f32_16x16x4_f32(
            false, Aop, false, Bop, (short)0, Zc, false, false);

        // Pack: key = (bits(max(d2,0)) & ~0x7FF) | candidate_index  (u32-monotonic in d2).
        // Clamp negatives (self-distance cancellation) via integer max: for finite floats,
        // bits-as-int < 0  <=>  value < 0, and max((int)bits,0) == bits(max(val,+0)).
        unsigned keys[8];
#pragma unroll
        for (int v = 0; v < 8; ++v) {
            unsigned jb = (unsigned)(j0 + v + (hi << 3));
            int db = (int)__float_as_uint(D[v]);
            unsigned cb = (unsigned)(db > 0 ? db : 0);   // one v_max_i32
            keys[v] = (cb & KEY_MASK) | jb;
        }
        // One reject test per tile for the whole 8-candidate strip.
        unsigned kmin = min(min(min(keys[0], keys[1]), min(keys[2], keys[3])),
                            min(min(keys[4], keys[5]), min(keys[6], keys[7])));
        if (kmin < topK[KNN - 1]) {
#pragma unroll
            for (int v = 0; v < 8; ++v) {
                unsigned kv = keys[v];
                if (kv < topK[KNN - 1]) {
                    topK[KNN - 1] = kv;
#pragma unroll
                    for (int k = KNN - 1; k > 0; --k) {
                        unsigned a = topK[k - 1], b = topK[k];
                        bool sw = b < a;
                        topK[k - 1] = sw ? b : a;
                        topK[k]     = sw ? a : b;
                    }
                }
            }
        }
    }

    // ---------------- Phase 2: spill keys (aliasing quads), merge halves, output ----------------
    __syncthreads();   // all waves done reading quads
    unsigned* topKL = (unsigned*)smemRaw;   // [8][2][16][10]
    {
        const int base = (((wave << 1) + hi) * 16 + col) * KNN;
#pragma unroll
        for (int k = 0; k < KNN; ++k) topKL[base + k] = topK[k];
    }
    __syncthreads();

    if (t < 128) {
        const int w2 = t >> 4;
        const int ba = (((w2 << 1) + 0) * 16 + (t & 15)) * KNN;
        const int bb = (((w2 << 1) + 1) * 16 + (t & 15)) * KNN;
        int pa = 0, pb = 0;
        float m0 = 0.f, m1 = 0.f, m2 = 0.f;
        float X0 = -3.4e38f, X1 = -3.4e38f, X2 = -3.4e38f;
#pragma unroll
        for (int r = 0; r < KNN; ++r) {
            unsigned ka = (pa < KNN) ? topKL[ba + pa] : 0xFFFFFFFFu;
            unsigned kb = (pb < KNN) ? topKL[bb + pb] : 0xFFFFFFFFu;
            bool ta = ka <= kb;
            unsigned kmin = ta ? ka : kb;
            pa += ta ? 1 : 0;
            pb += ta ? 0 : 1;
            int   j  = (int)(kmin & IDX_MASK);
            float d2 = __uint_as_float(kmin & KEY_MASK);
            float wgt = __expf(-10.0f * d2);
            float g0 = wgt * hL[j * 3 + 0];
            float g1 = wgt * hL[j * 3 + 1];
            float g2 = wgt * hL[j * 3 + 2];
            m0 += g0; m1 += g1; m2 += g2;
            X0 = fmaxf(X0, g0); X1 = fmaxf(X1, g1); X2 = fmaxf(X2, g2);
        }
        m0 *= (1.0f / KNN); m1 *= (1.0f / KNN); m2 *= (1.0f / KNN);

        const int g = gbase + wgIn * 128 + t;
        const float* xp = x + (size_t)g * 3;
        float o = bo[0];
        o = fmaf(xp[0], Wo[0], o);
        o = fmaf(xp[1], Wo[1], o);
        o = fmaf(xp[2], Wo[2], o);
        o = fmaf(m0, Wo[3], o);
        o = fmaf(m1, Wo[4], o);
        o = fmaf(m2, Wo[5], o);
        o = fmaf(X0, Wo[6], o);
        o = fmaf(X1, Wo[7], o);
        o = fmaf(X2, Wo[8], o);
        out[g] = o;
    }
}

extern "C" void kernel_launch(void* const* d_in, const int* in_sizes, int n_in,
                              void* d_out, int out_size, void* d_ws, size_t ws_size,
                              hipStream_t stream) {
    const float* x  = (const float*)d_in[0];
    const float* Ws = (const float*)d_in[1];
    const float* bs = (const float*)d_in[2];
    const float* Wh = (const float*)d_in[3];
    const float* bh = (const float*)d_in[4];
    const float* Wo = (const float*)d_in[5];
    const float* bo = (const float*)d_in[6];
    float* out = (float*)d_out;
    (void)in_sizes; (void)n_in; (void)out_size; (void)d_ws; (void)ws_size;

    dim3 grid(B_GRAPHS * WGS_PER_GRAPH);   // 1024 workgroups
    dim3 block(THREADS);
    gravnet_knn_kernel<<<grid, block, 0, stream>>>(x, Ws, bs, Wh, bh, Wo, bo, out);
}